// CenterPointDecoder_82884278879165
// MI455X (gfx1250) — compile-verified
//
#include <hip/hip_runtime.h>
#include <hip/hip_bf16.h>
#include <math.h>

#define HH 512
#define WW 512
#define HWSZ (HH * WW)
#define TOPK 50
#define NSLICE 32            // 2 tasks * 8 batch * 2 classes
#define BLKS_PER_SLICE 16
#define ROWS_PER_BLK 32
#define T_OUT 8              // output rows per tile
#define T_ROWS 10            // loaded rows per tile (with halo)
#define NTILES (ROWS_PER_BLK / T_OUT)
#define CAND_MAX 2560
#define NTHR 256
#define SCORE_THR 0.1f

typedef unsigned int uint32;
typedef unsigned long long u64;

typedef unsigned int u32x4 __attribute__((ext_vector_type(4)));
typedef int i32x4 __attribute__((ext_vector_type(4)));
typedef int i32x8 __attribute__((ext_vector_type(8)));

#if defined(__gfx1250__) && __has_builtin(__builtin_amdgcn_tensor_load_to_lds)
#define USE_TDM 1
#else
#define USE_TDM 0
#endif

// ---------------------------------------------------------------------------
// TDM: issue an async load of a (rows x 512) fp32 tile from global into LDS.
// Descriptor per CDNA5 ISA 8.3/8.4 (2D tensor, groups 2/3 zero). Caller
// synchronizes with s_wait_tensorcnt.
// ---------------------------------------------------------------------------
__device__ __forceinline__ void tdm_tile_issue(u64 gaddr, uint32 lds_addr, int rows) {
#if USE_TDM
  u32x4 g0;
  g0[0] = 1u;                                              // count=1, user D#
  g0[1] = lds_addr;                                        // LDS byte address
  g0[2] = (uint32)(gaddr & 0xFFFFFFFFull);                 // global_addr lo
  g0[3] = ((uint32)(gaddr >> 32) & 0x01FFFFFFu) | (2u << 30); // addr hi | type=2
  i32x8 g1;
  g1[0] = (2 << 16);        // workgroup_mask=0 (not in cluster), data_size=4B
  g1[1] = (WW << 16);       // tensor_dim0[15:0] = 512
  g1[2] = (rows << 16);     // tensor_dim0 hi=0 | tensor_dim1[15:0] = rows
  g1[3] = (WW << 16);       // tensor_dim1 hi=0 | tile_dim0 = 512
  g1[4] = rows;             // tile_dim1 = rows | tile_dim2 = 0
  g1[5] = WW;               // tensor_dim0_stride lo = 512 elements
  g1[6] = 0;
  g1[7] = 0;
  i32x4 gz = {0, 0, 0, 0};
#if defined(__clang_major__) && (__clang_major__ >= 23)
  i32x8 gz8 = {0, 0, 0, 0, 0, 0, 0, 0};
  __builtin_amdgcn_tensor_load_to_lds(g0, g1, gz, gz, gz8, 0);
#else
  __builtin_amdgcn_tensor_load_to_lds(g0, g1, gz, gz, 0);
#endif
#endif
}

__device__ __forceinline__ float stable_sigmoid(float c) {
  const float e = expf(-fabsf(c));
  return (c >= 0.0f) ? 1.0f / (1.0f + e) : e / (1.0f + e);
}

// ---------------------------------------------------------------------------
// K1: per (slice, 32-row block): sigmoid + 3x3 peak NMS + exact block top-50.
// TDM double-buffered: wave 0 issues tile k+1 into the alternate LDS buffer,
// all waves compute tile k concurrently; s_wait_tensorcnt 0 at loop top.
// Key = (score_bits << 32) | (0xFFFFFFFF - flat_idx): descending score,
// ascending index on ties (matches jax top_k tie-break).
// ---------------------------------------------------------------------------
__global__ __launch_bounds__(NTHR) void k1_peaks_topk(
    const float* __restrict__ hm0, const float* __restrict__ hm1,
    u64* __restrict__ blockTop) {
  __shared__ float tileA[T_ROWS * WW];  // 20 KB each; slot s <-> row r0-1+s
  __shared__ float tileB[T_ROWS * WW];
  __shared__ u64 cand[CAND_MAX];        // 20 KB
  __shared__ u64 red[NTHR];             // 2 KB
  __shared__ int ncand;

  const int tid = threadIdx.x;
  const int gb = blockIdx.x;                   // 0..511
  const int slice = gb / BLKS_PER_SLICE;       // 0..31
  const int blk = gb % BLKS_PER_SLICE;         // 0..15
  const int t = slice >> 4;
  const int bc = slice & 15;                   // b*2 + c
  const float* base = (t ? hm1 : hm0) + (size_t)bc * HWSZ;
  const bool wave0 = (tid >> 5) == 0;

  if (tid == 0) ncand = 0;

#if USE_TDM
  const uint32 ldsA = (uint32)(size_t)(&tileA[0]);  // LDS byte offsets
  const uint32 ldsB = (uint32)(size_t)(&tileB[0]);
  // Preload tile 0 into buffer A.
  {
    const int r0 = blk * ROWS_PER_BLK;
    const int rs = (r0 == 0) ? 0 : r0 - 1;
    const int re = (r0 + T_OUT + 1 > HH) ? HH : r0 + T_OUT + 1;
    if (wave0)
      tdm_tile_issue((u64)(size_t)(base + (size_t)rs * WW),
                     ldsA + (r0 == 0 ? (uint32)(WW * 4) : 0u), re - rs);
  }
#endif
  __syncthreads();  // ncand visible (and orders the preload issue)

  for (int tk = 0; tk < NTILES; ++tk) {
    const int r0 = blk * ROWS_PER_BLK + tk * T_OUT;
    const float* buf = (tk & 1) ? tileB : tileA;
#if USE_TDM
    if (wave0) __builtin_amdgcn_s_wait_tensorcnt(0);  // tile tk arrived
    __syncthreads();  // LDS visible to all waves; all done with buf[tk-1 mod 2]
    if (tk + 1 < NTILES) {  // overlap DMA of tile tk+1 with compute of tile tk
      const int r1 = r0 + T_OUT;            // r1 >= 8, no top-edge case
      const int rs1 = r1 - 1;
      const int re1 = (r1 + T_OUT + 1 > HH) ? HH : r1 + T_OUT + 1;
      if (wave0)
        tdm_tile_issue((u64)(size_t)(base + (size_t)rs1 * WW),
                       ((tk + 1) & 1) ? ldsB : ldsA, re1 - rs1);
    }
#else
    __syncthreads();
    {
      const int rs = (r0 == 0) ? 0 : r0 - 1;
      const int re = (r0 + T_OUT + 1 > HH) ? HH : r0 + T_OUT + 1;
      float* dst = (float*)buf + (r0 == 0 ? WW : 0);
      const float* src = base + (size_t)rs * WW;
      for (int i = tid; i < (re - rs) * WW; i += NTHR) dst[i] = src[i];
    }
    __syncthreads();
#endif

    for (int i = 0; i < (T_OUT * WW) / NTHR; ++i) {
      const int p = tid + i * NTHR;
      const int y = r0 + (p >> 9);
      const int x = p & (WW - 1);
      const int ly = y - r0 + 1;  // slot of row y (slot 0 <-> row r0-1)
      const float c = buf[ly * WW + x];
      const bool xm = x > 0, xp = x < WW - 1, ym = y > 0, yp = y < HH - 1;
      bool peak = true;  // OOB neighbors are -inf in the reference padding
      if (xm) peak = peak && (c >= buf[ly * WW + x - 1]);
      if (xp) peak = peak && (c >= buf[ly * WW + x + 1]);
      if (ym) {
        peak = peak && (c >= buf[(ly - 1) * WW + x]);
        if (xm) peak = peak && (c >= buf[(ly - 1) * WW + x - 1]);
        if (xp) peak = peak && (c >= buf[(ly - 1) * WW + x + 1]);
      }
      if (yp) {
        peak = peak && (c >= buf[(ly + 1) * WW + x]);
        if (xm) peak = peak && (c >= buf[(ly + 1) * WW + x - 1]);
        if (xp) peak = peak && (c >= buf[(ly + 1) * WW + x + 1]);
      }
      if (peak) {
        const float sig = stable_sigmoid(c);
        if (sig > SCORE_THR) {  // sub-threshold rows are zeroed -> tail pad
          const int flat = y * WW + x;
          const u64 key = ((u64)__float_as_uint(sig) << 32) |
                          (u64)(0xFFFFFFFFu - (uint32)flat);
          const int idx = atomicAdd(&ncand, 1);  // ds_add_rtn_u32
          if (idx < CAND_MAX) cand[idx] = key;
        }
      }
    }
  }
  __syncthreads();
  int n = ncand;
  if (n > CAND_MAX) n = CAND_MAX;

  // 50 rounds of exact block-wide max selection (keys are unique).
  for (int k = 0; k < TOPK; ++k) {
    u64 lmax = 0;
    int lpos = -1;
    for (int i2 = tid; i2 < n; i2 += NTHR)
      if (cand[i2] > lmax) { lmax = cand[i2]; lpos = i2; }
    red[tid] = lmax;
    __syncthreads();
    for (int s = NTHR / 2; s > 0; s >>= 1) {
      if (tid < s && red[tid + s] > red[tid]) red[tid] = red[tid + s];
      __syncthreads();
    }
    const u64 m = red[0];
    if (m != 0 && lmax == m && lpos >= 0) cand[lpos] = 0;  // unique winner
    if (tid == 0) blockTop[(size_t)gb * TOPK + k] = m;
    __syncthreads();
  }
}

// ---------------------------------------------------------------------------
// K2: per slice, merge 16 block lists (800 keys) -> exact sorted top-50.
// ---------------------------------------------------------------------------
__global__ __launch_bounds__(NTHR) void k2_merge_slice(
    const u64* __restrict__ blockTop, u64* __restrict__ sliceTop) {
  __shared__ u64 c2[BLKS_PER_SLICE * TOPK];
  __shared__ u64 red[NTHR];
  const int tid = threadIdx.x;
  const int slice = blockIdx.x;
  for (int i = tid; i < BLKS_PER_SLICE * TOPK; i += NTHR)
    c2[i] = blockTop[(size_t)slice * BLKS_PER_SLICE * TOPK + i];
  __syncthreads();
  for (int k = 0; k < TOPK; ++k) {
    u64 lmax = 0;
    int lpos = -1;
    for (int i2 = tid; i2 < BLKS_PER_SLICE * TOPK; i2 += NTHR)
      if (c2[i2] > lmax) { lmax = c2[i2]; lpos = i2; }
    red[tid] = lmax;
    __syncthreads();
    for (int s = NTHR / 2; s > 0; s >>= 1) {
      if (tid < s && red[tid + s] > red[tid]) red[tid] = red[tid + s];
      __syncthreads();
    }
    const u64 m = red[0];
    if (m != 0 && lmax == m && lpos >= 0) c2[lpos] = 0;
    if (tid == 0) sliceTop[(size_t)slice * TOPK + k] = m;
    __syncthreads();
  }
}

// ---------------------------------------------------------------------------
// K3: per (task, batch): merge the two class lists (ties -> class 0, matching
// concat-index tie-break), gather bbox channels, decode, mask, write output.
// ---------------------------------------------------------------------------
__global__ __launch_bounds__(64) void k3_decode(
    const float* __restrict__ reg0, const float* __restrict__ hei0,
    const float* __restrict__ dim0, const float* __restrict__ rot0,
    const float* __restrict__ vel0,
    const float* __restrict__ reg1, const float* __restrict__ hei1,
    const float* __restrict__ dim1, const float* __restrict__ rot1,
    const float* __restrict__ vel1,
    const u64* __restrict__ sliceTop, float* __restrict__ out) {
  __shared__ u64 l0[TOPK], l1[TOPK], mk[TOPK];
  __shared__ int mcls[TOPK];
  const int tid = threadIdx.x;
  const int t = blockIdx.x >> 3;  // task
  const int b = blockIdx.x & 7;   // batch
  const int s0 = t * 16 + b * 2;
  if (tid < TOPK) {
    l0[tid] = sliceTop[(size_t)s0 * TOPK + tid];
    l1[tid] = sliceTop[(size_t)(s0 + 1) * TOPK + tid];
  }
  __syncthreads();
  if (tid == 0) {  // sequential 50-step merge of two sorted lists
    int i = 0, j = 0;
    for (int r = 0; r < TOPK; ++r) {
      const uint32 h0 = (i < TOPK) ? (uint32)(l0[i] >> 32) : 0u;
      const uint32 h1 = (j < TOPK) ? (uint32)(l1[j] >> 32) : 0u;
      const bool take0 = (j >= TOPK) || (i < TOPK && h0 >= h1);
      if (take0) { mk[r] = l0[i]; mcls[r] = 0; ++i; }
      else       { mk[r] = l1[j]; mcls[r] = 1; ++j; }
    }
  }
  __syncthreads();
  if (tid < TOPK) {
    const u64 key = mk[tid];
    const float score = __uint_as_float((uint32)(key >> 32));
    const bool has = score > SCORE_THR;
    const int flat = has ? (int)(0xFFFFFFFFu - (uint32)key) : 0;
    const float xs = (float)(flat & (WW - 1));
    const float ys = (float)(flat >> 9);
    const float* regp = (t ? reg1 : reg0) + (size_t)b * 2 * HWSZ;
    const float* heip = (t ? hei1 : hei0) + (size_t)b * HWSZ;
    const float* dimp = (t ? dim1 : dim0) + (size_t)b * 3 * HWSZ;
    const float* rotp = (t ? rot1 : rot0) + (size_t)b * 2 * HWSZ;
    const float* velp = (t ? vel1 : vel0) + (size_t)b * 2 * HWSZ;
    const float x = (xs + regp[flat]) * 4.0f;
    const float y = (ys + regp[HWSZ + flat]) * 4.0f;
    const float he = heip[flat];
    const float d0 = expf(dimp[flat]);
    const float d1 = expf(dimp[HWSZ + flat]);
    const float d2 = expf(dimp[2 * HWSZ + flat]);
    const float ra = atan2f(rotp[flat], rotp[HWSZ + flat]);
    const float v0 = velp[flat];
    const float v1 = velp[HWSZ + flat];
    const bool mask = has && (x > 0.0f) && (x < 2048.0f) &&
                      (y > 0.0f) && (y < 2048.0f);
    const float mm = mask ? 1.0f : 0.0f;
    const float gcls = (float)(mcls[tid] + (t ? 2 : 0));
    const size_t rowbase = ((size_t)b * 100 + (size_t)t * TOPK + tid) * 11;
    out[rowbase + 0] = x * mm;
    out[rowbase + 1] = y * mm;
    out[rowbase + 2] = he * mm;
    out[rowbase + 3] = d0 * mm;
    out[rowbase + 4] = d1 * mm;
    out[rowbase + 5] = d2 * mm;
    out[rowbase + 6] = ra * mm;
    out[rowbase + 7] = v0 * mm;
    out[rowbase + 8] = v1 * mm;
    out[rowbase + 9] = score * mm;
    out[rowbase + 10] = gcls * mm;
    out[8 * 100 * 11 + (size_t)b * 100 + (size_t)t * TOPK + tid] = mm;
  }
}

extern "C" void kernel_launch(void* const* d_in, const int* in_sizes, int n_in,
                              void* d_out, int out_size, void* d_ws,
                              size_t ws_size, hipStream_t stream) {
  (void)in_sizes; (void)n_in; (void)out_size; (void)ws_size;
  const float* hm0 = (const float*)d_in[0];
  const float* reg0 = (const float*)d_in[1];
  const float* hei0 = (const float*)d_in[2];
  const float* dim0 = (const float*)d_in[3];
  const float* rot0 = (const float*)d_in[4];
  const float* vel0 = (const float*)d_in[5];
  const float* hm1 = (const float*)d_in[6];
  const float* reg1 = (const float*)d_in[7];
  const float* hei1 = (const float*)d_in[8];
  const float* dim1 = (const float*)d_in[9];
  const float* rot1 = (const float*)d_in[10];
  const float* vel1 = (const float*)d_in[11];
  float* out = (float*)d_out;

  u64* blockTop = (u64*)d_ws;                               // 512*50 u64
  u64* sliceTop = blockTop + (size_t)NSLICE * BLKS_PER_SLICE * TOPK;  // 32*50

  k1_peaks_topk<<<NSLICE * BLKS_PER_SLICE, NTHR, 0, stream>>>(hm0, hm1, blockTop);
  k2_merge_slice<<<NSLICE, NTHR, 0, stream>>>(blockTop, sliceTop);
  k3_decode<<<16, 64, 0, stream>>>(reg0, hei0, dim0, rot0, vel0,
                                   reg1, hei1, dim1, rot1, vel1,
                                   sliceTop, out);
}